// PairwiseSubtractionLayer_23665269801120
// MI455X (gfx1250) — compile-verified
//
#include <hip/hip_runtime.h>
#include <cstdint>

// Problem dims fixed by the reference: N=64, L=512, F=2048, slope=10.
#define N_  64
#define L_  512
#define F_  2048
#define SLOPE_ 10.0f

// ---------------------------------------------------------------------------
// Kernel 1: zero the per-n running max scratch (d_ws is poisoned by harness).
// ---------------------------------------------------------------------------
__global__ void pw_zero_ws(unsigned* __restrict__ ws) {
    ws[threadIdx.x] = 0u;
}

// ---------------------------------------------------------------------------
// Kernel 2: per (n, l) nearest-neighbor Chebyshev distance against the 2048
// B points, then block-max over l, then global atomic max into ws[n].
//
// Block = 256 threads (8 wave32), grid = (N, L/64).
//   tid = fchunk*64 + lidx : 64 l-values x 4 f-chunks of 512.
// B (2*2048 f32 = 16KB) is staged into LDS with CDNA5 async-to-LDS loads
// (ASYNCcnt) before the compute loop.
// ---------------------------------------------------------------------------
__global__ __launch_bounds__(256)
void pw_cheb_nn_kernel(const float* __restrict__ A,
                       const float* __restrict__ B,
                       unsigned* __restrict__ ws) {
    __shared__ float lB[2 * F_];     // lB[f] = B[0][f], lB[F_+f] = B[1][f]
    __shared__ float red[256];
    __shared__ float red2[64];

    const int tid    = threadIdx.x;
    const int n      = blockIdx.x;
    const int lbase  = blockIdx.y * 64;
    const int lidx   = tid & 63;
    const int fchunk = tid >> 6;

    // ---- Async stage B -> LDS: 4 rounds x 256 lanes x 16 bytes = 16KB ----
    {
        const unsigned lbaseaddr = (unsigned)(uintptr_t)(&lB[0]);
        #pragma unroll
        for (int i = 0; i < 4; ++i) {
            const unsigned byteoff = (unsigned)(tid * 16 + i * 4096);
            unsigned       lofs    = lbaseaddr + byteoff;
            uint64_t       gaddr   = (uint64_t)(uintptr_t)((const char*)B + byteoff);
            // VDST = LDS byte address (per-lane VGPR), VADDR = 64-bit global addr.
            asm volatile("global_load_async_to_lds_b128 %0, %1, off"
                         :: "v"(lofs), "v"(gaddr)
                         : "memory");
        }
#if __has_builtin(__builtin_amdgcn_s_wait_asynccnt)
        __builtin_amdgcn_s_wait_asynccnt(0);
#else
        asm volatile("s_wait_asynccnt 0x0" ::: "memory");
#endif
    }
    __syncthreads();

    // ---- Per-thread: min over 512 f of max(|a0-b0|, |a1-b1|) ----
    const size_t abase = ((size_t)n * L_ + (size_t)(lbase + lidx)) * 2;
    const float a0 = A[abase + 0];
    const float a1 = A[abase + 1];

    float dmin = 3.402823466e38f;
    const int f0 = fchunk * (F_ / 4);
    #pragma unroll 8
    for (int f = f0; f < f0 + (F_ / 4); ++f) {
        const float b0 = lB[f];
        const float b1 = lB[F_ + f];
        const float d  = fmaxf(fabsf(a0 - b0), fabsf(a1 - b1));
        dmin = fminf(dmin, d);
    }
    red[tid] = dmin;
    __syncthreads();

    // ---- Combine the 4 f-chunks per l ----
    if (tid < 64) {
        red2[tid] = fminf(fminf(red[tid],       red[tid + 64]),
                          fminf(red[tid + 128], red[tid + 192]));
    }
    __syncthreads();

    // ---- Block max over the 64 l-values: one wave32 + xor shuffles ----
    if (tid < 32) {
        float m = fmaxf(red2[tid], red2[tid + 32]);
        #pragma unroll
        for (int off = 16; off > 0; off >>= 1)
            m = fmaxf(m, __shfl_xor(m, off, 32));
        if (tid == 0) {
            // d >= 0, so IEEE order == unsigned-int order on the bit pattern.
            atomicMax(&ws[n], __float_as_uint(m));
        }
    }
}

// ---------------------------------------------------------------------------
// Kernel 3: out[n] = g(D) = sigmoid(10 D) * sigmoid(-10 D)
//                  = e / (1+e)^2 with e = exp(-10 D), D >= 0 (stable form).
// ---------------------------------------------------------------------------
__global__ void pw_finalize(const unsigned* __restrict__ ws,
                            float* __restrict__ out) {
    const int i  = threadIdx.x;
    const float D = __uint_as_float(ws[i]);
    const float e = expf(-SLOPE_ * D);
    const float t = 1.0f + e;
    out[i] = e / (t * t);
}

// ---------------------------------------------------------------------------
extern "C" void kernel_launch(void* const* d_in, const int* in_sizes, int n_in,
                              void* d_out, int out_size, void* d_ws, size_t ws_size,
                              hipStream_t stream) {
    (void)in_sizes; (void)n_in; (void)out_size; (void)ws_size;
    const float* A  = (const float*)d_in[0];   // (64, 512, 2) f32
    const float* B  = (const float*)d_in[1];   // (1, 2, 2048) f32
    float*       out = (float*)d_out;          // (64, 1) f32
    unsigned*    ws  = (unsigned*)d_ws;        // 64 running maxima (bits of f32>=0)

    pw_zero_ws<<<1, N_, 0, stream>>>(ws);
    pw_cheb_nn_kernel<<<dim3(N_, L_ / 64), 256, 0, stream>>>(A, B, ws);
    pw_finalize<<<1, N_, 0, stream>>>(ws, out);
}